// STHCW_17446157156967
// MI455X (gfx1250) — compile-verified
//
#include <hip/hip_runtime.h>
#include <math.h>
#include <stdint.h>

#define EMB 32

typedef float v2f __attribute__((ext_vector_type(2)));
typedef float v8f __attribute__((ext_vector_type(8)));

// ---------------------------------------------------------------- zero fill
__global__ void zero_f4(float4* __restrict__ p, int n4) {
    int i = blockIdx.x * blockDim.x + threadIdx.x;
    if (i < n4) p[i] = make_float4(0.f, 0.f, 0.f, 0.f);
}

// ---------------------------------------------------------------- SpMM scatter
// One wave handles 32 edges. Lane = embedding column (EMB == wave32 width).
// Edge data (rows/cols/vals, 3 x 128B per 32 edges) is staged into the wave's
// private LDS slice with CDNA5 async copies (GLOBAL_LOAD_ASYNC_TO_LDS_B32,
// tracked by ASYNCcnt), then consumed via broadcast DS reads. The gather of
// X[col] is one 128B line per edge; the scatter-add into Y[row] is a
// coalesced 32-lane f32 atomic (everything L2-resident on MI455X's 192MB L2).
__global__ void spmm_kernel(const float* __restrict__ vals,
                            const int*   __restrict__ rows,
                            const int*   __restrict__ cols,
                            const float* __restrict__ X,
                            float*       __restrict__ Y,
                            int nnz) {
    __shared__ int   s_rows[8][32];   // 8 waves per 256-thread block
    __shared__ int   s_cols[8][32];
    __shared__ float s_vals[8][32];

    const int lane = threadIdx.x & 31;
    const int w    = threadIdx.x >> 5;
    const int wave = (blockIdx.x * blockDim.x + threadIdx.x) >> 5;
    const int base = wave * 32;
    if (base >= nnz) return;

    // Clamp tail lanes onto the last valid edge (extra copies are never read:
    // the consume loop is bounded by cnt).
    const int e = (base + lane < nnz) ? (base + lane) : (nnz - 1);

    // LDS byte addresses: hardware maps flat->LDS via addr[31:0].
    uint32_t lr = (uint32_t)(uintptr_t)&s_rows[w][lane];
    uint32_t lc = (uint32_t)(uintptr_t)&s_cols[w][lane];
    uint32_t lv = (uint32_t)(uintptr_t)&s_vals[w][lane];
    const int*   pr = rows + e;
    const int*   pc = cols + e;
    const float* pv = vals + e;

    // Async line-sized copies into LDS (ASYNCcnt), then wait for completion.
    asm volatile("global_load_async_to_lds_b32 %0, %1, off"
                 :: "v"(lr), "v"(pr) : "memory");
    asm volatile("global_load_async_to_lds_b32 %0, %1, off"
                 :: "v"(lc), "v"(pc) : "memory");
    asm volatile("global_load_async_to_lds_b32 %0, %1, off"
                 :: "v"(lv), "v"(pv) : "memory");
    asm volatile("s_wait_asynccnt 0x0" ::: "memory");

    const int cnt = (nnz - base < 32) ? (nnz - base) : 32;
    #pragma unroll 4
    for (int j = 0; j < cnt; ++j) {
        const int   rj = s_rows[w][j];   // uniform-address broadcast DS reads
        const int   cj = s_cols[w][j];
        const float vj = s_vals[w][j];
        const float x  = X[cj * EMB + lane];
        atomicAdd(&Y[rj * EMB + lane], vj * x);
    }
}

// ---------------------------------------------------------------- softmax(alpha)
__device__ __forceinline__ void softmax4(const float* __restrict__ alpha, int Ka,
                                         float a[4]) {
    float t[4];
    float m = -3.4e38f;
    #pragma unroll
    for (int i = 0; i < 4; ++i) {
        t[i] = (i < Ka) ? alpha[i] : -3.4e38f;
        m = fmaxf(m, t[i]);
    }
    float s = 0.f;
    #pragma unroll
    for (int i = 0; i < 4; ++i) { t[i] = __expf(t[i] - m); s += t[i]; }
    const float inv = 1.0f / s;
    #pragma unroll
    for (int i = 0; i < 4; ++i) a[i] = t[i] * inv;
}

// ---------------------------------------------------------------- WMMA combine
// out[c] = sum_k a[k] * E_k[c], contraction over k=4 mapped onto
// V_WMMA_F32_16X16X4_F32:  D(16x16) = A(16x4) x B(4x16)
//   A[M=i, K=k] = E_k[base + i]   (16 flat output elements per wave)
//   B[K=k, N=j] = a[k]            (replicated across columns)
// => D[i, j] = out[base+i] for every j.
// f32 A 16x4 layout (ISA 7.12.2): lanes 0-15 -> M=lane, VGPR0=K0/VGPR1=K1;
//                                 lanes 16-31 -> M=lane-16, VGPR0=K2/VGPR1=K3.
// D 16x16 f32 layout: lane N, VGPR p -> (M=p, N) lanes 0-15; (M=p+8) lanes 16-31.
__global__ void combine_wmma_kernel(const float* __restrict__ E0,
                                    const float* __restrict__ E1,
                                    const float* __restrict__ E2,
                                    const float* __restrict__ E3,
                                    const float* __restrict__ alpha,
                                    float* __restrict__ out,
                                    int total, int Ka) {
    float a[4];
    softmax4(alpha, Ka, a);

    const int tid  = blockIdx.x * blockDim.x + threadIdx.x;
    const int wave = tid >> 5;
    const int lane = tid & 31;
    const int base = wave * 16;
    if (base >= total) return;

    if (base + 16 <= total) {
        const bool lower = lane < 16;
        const int  idx   = base + (lane & 15);

        v2f A;
        A.x = lower ? E0[idx] : E2[idx];
        A.y = lower ? E1[idx] : E3[idx];

        v2f B;
        B.x = lower ? a[0] : a[2];
        B.y = lower ? a[1] : a[3];

        v8f C = {};
        v8f D = __builtin_amdgcn_wmma_f32_16x16x4_f32(
            /*neg_a=*/false, A, /*neg_b=*/false, B,
            /*c_mod=*/(short)0, C, /*reuse_a=*/false, /*reuse_b=*/false);

        // Lane 0 holds M=0..7 in D[0..7]; lane 16 holds M=8..15.
        if (lane == 0) {
            float4 lo = make_float4(D[0], D[1], D[2], D[3]);
            float4 hi = make_float4(D[4], D[5], D[6], D[7]);
            *(float4*)(out + base)     = lo;
            *(float4*)(out + base + 4) = hi;
        } else if (lane == 16) {
            float4 lo = make_float4(D[0], D[1], D[2], D[3]);
            float4 hi = make_float4(D[4], D[5], D[6], D[7]);
            *(float4*)(out + base + 8)  = lo;
            *(float4*)(out + base + 12) = hi;
        }
    } else {
        // ragged tail (not hit when total % 16 == 0): scalar path
        const int i = base + (lane & 15);
        if (lane < 16 && i < total) {
            out[i] = a[0] * E0[i] + a[1] * E1[i] + a[2] * E2[i] + a[3] * E3[i];
        }
    }
}

// ---------------------------------------------------------------- scalar fallback
__global__ void axpy_kernel(const float* __restrict__ E,
                            const float* __restrict__ alpha, int Ka, int k,
                            float* __restrict__ out, int total, int accum) {
    int i = blockIdx.x * blockDim.x + threadIdx.x;
    if (i >= total) return;
    float a[4];
    softmax4(alpha, Ka, a);
    float r = a[k] * E[i];
    if (accum) r += out[i];
    out[i] = r;
}

// ---------------------------------------------------------------- host launch
extern "C" void kernel_launch(void* const* d_in, const int* in_sizes, int n_in,
                              void* d_out, int out_size, void* d_ws, size_t ws_size,
                              hipStream_t stream) {
    const float* A_vals = (const float*)d_in[0];
    const float* W0     = (const float*)d_in[1];
    const float* alpha  = (const float*)d_in[2];
    const int*   A_rows = (const int*)d_in[3];
    const int*   A_cols = (const int*)d_in[4];

    const int nnz   = in_sizes[0];
    const int total = in_sizes[1];   // N * EMB
    const int Ka    = in_sizes[2];   // K + 1  (== 4 in the reference)
    float* out = (float*)d_out;

    const size_t bufBytes = (size_t)total * sizeof(float);
    const int n4         = total / 4;
    const int zeroBlocks = (n4 + 255) / 256;
    const int waves      = (nnz + 31) / 32;
    const int spmmBlocks = (waves * 32 + 255) / 256;

    if (Ka == 4 && ws_size >= 4 * bufBytes) {
        // Four hop buffers resident in d_ws (all L2-resident at 25.6MB each).
        float* E[4];
        for (int i = 0; i < 4; ++i)
            E[i] = (float*)((char*)d_ws + (size_t)i * bufBytes);

        const float* src = W0;
        for (int k = 0; k < 4; ++k) {   // E0 = A@W0; E_{k} = A@E_{k-1}
            zero_f4<<<zeroBlocks, 256, 0, stream>>>((float4*)E[k], n4);
            spmm_kernel<<<spmmBlocks, 256, 0, stream>>>(A_vals, A_rows, A_cols,
                                                        src, E[k], nnz);
            src = E[k];
        }

        const int cwaves  = (total + 15) / 16;
        const int cBlocks = (cwaves * 32 + 255) / 256;
        combine_wmma_kernel<<<cBlocks, 256, 0, stream>>>(E[0], E[1], E[2], E[3],
                                                         alpha, out, total, Ka);
    } else {
        // Ping-pong fallback (2 buffers), scalar weighted accumulation.
        float* B0 = (float*)d_ws;
        float* B1 = (float*)((char*)d_ws + bufBytes);
        const int hops    = (Ka < 1) ? 1 : ((Ka > 4) ? 4 : Ka);
        const int tBlocks = (total + 255) / 256;
        const float* src = W0;
        for (int k = 0; k < hops; ++k) {
            float* dst = (k & 1) ? B1 : B0;
            zero_f4<<<zeroBlocks, 256, 0, stream>>>((float4*)dst, n4);
            spmm_kernel<<<spmmBlocks, 256, 0, stream>>>(A_vals, A_rows, A_cols,
                                                        src, dst, nnz);
            axpy_kernel<<<tBlocks, 256, 0, stream>>>(dst, alpha, Ka, k,
                                                     out, total, k > 0);
            src = dst;
        }
    }
}